// Evaluation_8349416424064
// MI455X (gfx1250) — compile-verified
//
#include <hip/hip_runtime.h>
#include <math.h>

typedef __attribute__((ext_vector_type(2))) float v2f;
typedef __attribute__((ext_vector_type(8))) float v8f;

#define B_ 2
#define V_ 4
#define C_ 64
#define G_ 8
#define H_ 64
#define W_ 80
#define D_ 48
#define NB_ 9
#define HW_ (H_*W_)        // 5120
#define PIX_ (B_*HW_)      // 10240
#define POS_ (B_*D_*HW_)   // 491520

// workspace layout in floats
#define WS_ROT   0
#define SIMV_STRIDE ((size_t)B_*G_*D_*HW_)          // 3,932,160 per view
#define WS_SIM   128
#define WS_Y0    (WS_SIM + (size_t)V_*SIMV_STRIDE)  // + 15,728,640
#define WS_SCORE (WS_Y0 + (size_t)POS_)

// ---------------------------------------------------------------- 4x4 inverse
__device__ inline void inv4x4(const float m[16], float inv[16]) {
  float i0 =  m[5]*m[10]*m[15]-m[5]*m[11]*m[14]-m[9]*m[6]*m[15]+m[9]*m[7]*m[14]+m[13]*m[6]*m[11]-m[13]*m[7]*m[10];
  float i4 = -m[4]*m[10]*m[15]+m[4]*m[11]*m[14]+m[8]*m[6]*m[15]-m[8]*m[7]*m[14]-m[12]*m[6]*m[11]+m[12]*m[7]*m[10];
  float i8 =  m[4]*m[ 9]*m[15]-m[4]*m[11]*m[13]-m[8]*m[5]*m[15]+m[8]*m[7]*m[13]+m[12]*m[5]*m[11]-m[12]*m[7]*m[ 9];
  float i12= -m[4]*m[ 9]*m[14]+m[4]*m[10]*m[13]+m[8]*m[5]*m[14]-m[8]*m[6]*m[13]-m[12]*m[5]*m[10]+m[12]*m[6]*m[ 9];
  float i1 = -m[1]*m[10]*m[15]+m[1]*m[11]*m[14]+m[9]*m[2]*m[15]-m[9]*m[3]*m[14]-m[13]*m[2]*m[11]+m[13]*m[3]*m[10];
  float i5 =  m[0]*m[10]*m[15]-m[0]*m[11]*m[14]-m[8]*m[2]*m[15]+m[8]*m[3]*m[14]+m[12]*m[2]*m[11]-m[12]*m[3]*m[10];
  float i9 = -m[0]*m[ 9]*m[15]+m[0]*m[11]*m[13]+m[8]*m[1]*m[15]-m[8]*m[3]*m[13]-m[12]*m[1]*m[11]+m[12]*m[3]*m[ 9];
  float i13=  m[0]*m[ 9]*m[14]-m[0]*m[10]*m[13]-m[8]*m[1]*m[14]+m[8]*m[2]*m[13]+m[12]*m[1]*m[10]-m[12]*m[2]*m[ 9];
  float i2 =  m[1]*m[ 6]*m[15]-m[1]*m[ 7]*m[14]-m[5]*m[2]*m[15]+m[5]*m[3]*m[14]+m[13]*m[2]*m[ 7]-m[13]*m[3]*m[ 6];
  float i6 = -m[0]*m[ 6]*m[15]+m[0]*m[ 7]*m[14]+m[4]*m[2]*m[15]-m[4]*m[3]*m[14]-m[12]*m[2]*m[ 7]+m[12]*m[3]*m[ 6];
  float i10=  m[0]*m[ 5]*m[15]-m[0]*m[ 7]*m[13]-m[4]*m[1]*m[15]+m[4]*m[3]*m[13]+m[12]*m[1]*m[ 7]-m[12]*m[3]*m[ 5];
  float i14= -m[0]*m[ 5]*m[14]+m[0]*m[ 6]*m[13]+m[4]*m[1]*m[14]-m[4]*m[2]*m[13]-m[12]*m[1]*m[ 6]+m[12]*m[2]*m[ 5];
  float i3 = -m[1]*m[ 6]*m[11]+m[1]*m[ 7]*m[10]+m[5]*m[2]*m[11]-m[5]*m[3]*m[10]-m[ 9]*m[2]*m[ 7]+m[ 9]*m[3]*m[ 6];
  float i7 =  m[0]*m[ 6]*m[11]-m[0]*m[ 7]*m[10]-m[4]*m[2]*m[11]+m[4]*m[3]*m[10]+m[ 8]*m[2]*m[ 7]-m[ 8]*m[3]*m[ 6];
  float i11= -m[0]*m[ 5]*m[11]+m[0]*m[ 7]*m[ 9]+m[4]*m[1]*m[11]-m[4]*m[3]*m[ 9]-m[ 8]*m[1]*m[ 7]+m[ 8]*m[3]*m[ 5];
  float i15=  m[0]*m[ 5]*m[10]-m[0]*m[ 6]*m[ 9]-m[4]*m[1]*m[10]+m[4]*m[2]*m[ 9]+m[ 8]*m[1]*m[ 6]-m[ 8]*m[2]*m[ 5];
  float det = m[0]*i0 + m[1]*i4 + m[2]*i8 + m[3]*i12;
  float r = 1.0f/det;
  inv[0]=i0*r;  inv[1]=i1*r;  inv[2]=i2*r;  inv[3]=i3*r;
  inv[4]=i4*r;  inv[5]=i5*r;  inv[6]=i6*r;  inv[7]=i7*r;
  inv[8]=i8*r;  inv[9]=i9*r;  inv[10]=i10*r; inv[11]=i11*r;
  inv[12]=i12*r; inv[13]=i13*r; inv[14]=i14*r; inv[15]=i15*r;
}

// -------------------------------------------------------- k0: rot/trans setup
__global__ void k_proj(const float* __restrict__ ref_proj,
                       const float* __restrict__ src_projs,
                       float* __restrict__ ws_rot) {
  int t = threadIdx.x;
  if (t >= V_*B_) return;
  int v = t / B_, b = t % B_;
  float m[16], inv[16];
  #pragma unroll
  for (int i = 0; i < 16; i++) m[i] = ref_proj[b*16 + i];
  inv4x4(m, inv);
  const float* S = src_projs + (v*B_ + b)*16;
  float* o = ws_rot + t*12;   // [0..8]=rot row-major, [9..11]=trans
  #pragma unroll
  for (int i = 0; i < 3; i++) {
    #pragma unroll
    for (int j = 0; j < 4; j++) {
      float acc = 0.f;
      #pragma unroll
      for (int k = 0; k < 4; k++) acc += S[i*4 + k] * inv[k*4 + j];
      if (j < 3) o[i*3 + j] = acc; else o[9 + i] = acc;
    }
  }
}

// --------------------------------- k1: fused warp + bilinear + group similarity
__global__ __launch_bounds__(256) void k_warp_sim(
    const float* __restrict__ ref_feat,   // (B,C,H,W)
    const float* __restrict__ src_feats,  // (V,B,C,H,W)
    const float* __restrict__ depth,      // (B,D,H,W)
    const float* __restrict__ ws_rot,
    float* __restrict__ sim_out) {        // (V,B,G,D,H,W)
  int idx = blockIdx.x * blockDim.x + threadIdx.x;  // [0, V*B*D*HW)
  int hw = idx % HW_;
  int r  = idx / HW_;
  int d  = r % D_; r /= D_;
  int b  = r % B_;
  int v  = r / B_;
  int w = hw % W_, h = hw / W_;

  const float* RT = ws_rot + (v*B_ + b)*12;
  float dep = depth[(size_t)(b*D_ + d)*HW_ + hw];
  float fx = (float)w, fy = (float)h;
  float rx = RT[0]*fx + RT[1]*fy + RT[2];
  float ry = RT[3]*fx + RT[4]*fy + RT[5];
  float rz = RT[6]*fx + RT[7]*fy + RT[8];
  float px = rx*dep + RT[9];
  float py = ry*dep + RT[10];
  float pz = rz*dep + RT[11];
  if (pz <= 0.001f) { px = (float)W_; py = (float)H_; pz = 1.0f; }
  // align_corners=True grid: ix = px/pz exactly
  float ix = px / pz, iy = py / pz;
  ix = fminf(fmaxf(ix, -1.0e4f), 1.0e4f);
  iy = fminf(fmaxf(iy, -1.0e4f), 1.0e4f);
  float x0f = floorf(ix), y0f = floorf(iy);
  float ax = ix - x0f, ay = iy - y0f;

  float wt[4]; int off[4];
  #pragma unroll
  for (int tap = 0; tap < 4; tap++) {
    int dx = tap & 1, dy = tap >> 1;
    float xi = x0f + (float)dx, yi = y0f + (float)dy;
    float wgt = (dx ? ax : 1.f - ax) * (dy ? ay : 1.f - ay);
    bool valid = (xi >= 0.f) && (xi <= (float)(W_-1)) && (yi >= 0.f) && (yi <= (float)(H_-1));
    wt[tap] = valid ? wgt : 0.f;
    int xc = min(max((int)xi, 0), W_-1);
    int yc = min(max((int)yi, 0), H_-1);
    off[tap] = yc*W_ + xc;
  }

  const float* src = src_feats + (size_t)(v*B_ + b)*C_*HW_;
  const float* ref = ref_feat + (size_t)b*C_*HW_ + hw;
  float simg[8];
  #pragma unroll
  for (int g = 0; g < 8; g++) {
    float a = 0.f;
    #pragma unroll
    for (int cc = 0; cc < 8; cc++) {
      int c = g*8 + cc;
      const float* sc = src + (size_t)c*HW_;
      float sval = wt[0]*sc[off[0]] + wt[1]*sc[off[1]] + wt[2]*sc[off[2]] + wt[3]*sc[off[3]];
      a += sval * ref[(size_t)c*HW_];
    }
    simg[g] = a * 0.125f;   // mean over C/G=8
  }
  float* so = sim_out + (size_t)(v*B_ + b)*G_*D_*HW_ + (size_t)d*HW_ + hw;
  #pragma unroll
  for (int g = 0; g < 8; g++) so[(size_t)g*D_*HW_] = simg[g];
}

// --------------------------- k2: pixelwise MLP (WMMA f32 16x16x4) + max-sigmoid
__global__ __launch_bounds__(256) void k_pixelwise(
    const float* __restrict__ sim,
    const float* __restrict__ w0, const float* __restrict__ s0, const float* __restrict__ b0,
    const float* __restrict__ w1, const float* __restrict__ s1, const float* __restrict__ b1,
    const float* __restrict__ w2, const float* __restrict__ b2,
    float* __restrict__ vw_out) {          // (B,V,H,W)
  __shared__ float lds[8][272];            // 16 rows * stride 17 per wave
  int lane  = threadIdx.x & 31;
  int wslot = threadIdx.x >> 5;
  int wid   = blockIdx.x*8 + wslot;        // [0, V * PIX/16)
  const int tilesPerView = PIX_/16;        // 640
  int v    = wid / tilesPerView;
  int tile = wid % tilesPerView;
  int b    = tile / (HW_/16);
  int m    = lane & 15;
  int hi   = lane >> 4;
  int hw   = (tile % (HW_/16))*16 + m;

  // A operands per ISA 16x4 f32 layout: lane=M, VGPR0<->K{0,2}, VGPR1<->K{1,3}
  v2f A0 = { w0[m*8 + 2*hi],     w0[m*8 + 1 + 2*hi] };
  v2f A1 = { w0[m*8 + 4 + 2*hi], w0[m*8 + 5 + 2*hi] };
  v2f A2[4];
  #pragma unroll
  for (int t = 0; t < 4; t++) {
    float axv = (m < 8) ? w1[m*16 + 4*t + 2*hi]     : 0.f;
    float ayv = (m < 8) ? w1[m*16 + 4*t + 1 + 2*hi] : 0.f;
    A2[t] = v2f{axv, ayv};
  }
  float sc0[8], bi0[8], sc1[8], bi1[8], w2v[8];
  #pragma unroll
  for (int j = 0; j < 8; j++) { sc0[j] = s0[j + 8*hi]; bi0[j] = b0[j + 8*hi]; }
  #pragma unroll
  for (int j = 0; j < 8; j++) { sc1[j] = s1[j]; bi1[j] = b1[j]; w2v[j] = w2[j]; }
  float bias2 = b2[0];

  const float* sbase = sim + (size_t)(v*B_ + b)*G_*D_*HW_ + hw;
  const size_t gs = (size_t)D_*HW_;
  float* myl = lds[wslot];
  float vmax = 0.f;   // sigmoid > 0

  for (int d = 0; d < D_; d++) {
    const float* sd = sbase + (size_t)d*HW_;
    // B operand: lane=N (pixel col), K striping matches A
    v2f B0 = { sd[(size_t)(2*hi    )*gs], sd[(size_t)(2*hi + 1)*gs] };
    v2f B1 = { sd[(size_t)(2*hi + 4)*gs], sd[(size_t)(2*hi + 5)*gs] };
    v8f c0 = {};
    c0 = __builtin_amdgcn_wmma_f32_16x16x4_f32(false, A0, false, B0, (short)0, c0, false, false);
    c0 = __builtin_amdgcn_wmma_f32_16x16x4_f32(false, A1, false, B1, (short)0, c0, false, false);
    // h0 = relu(c*s0 + b0); C/D layout: VGPR j -> row j+8*hi, col m
    #pragma unroll
    for (int j = 0; j < 8; j++) {
      float h0v = fmaxf(c0[j]*sc0[j] + bi0[j], 0.f);
      myl[(j + 8*hi)*17 + m] = h0v;
    }
    asm volatile("s_wait_dscnt 0" ::: "memory");
    v8f c1 = {};
    #pragma unroll
    for (int t = 0; t < 4; t++) {
      v2f Bt = { myl[(4*t + 2*hi)*17 + m], myl[(4*t + 1 + 2*hi)*17 + m] };
      c1 = __builtin_amdgcn_wmma_f32_16x16x4_f32(false, A2[t], false, Bt, (short)0, c1, false, false);
    }
    if (hi == 0) {  // rows 0..7 are the valid 8 outputs
      #pragma unroll
      for (int j = 0; j < 8; j++) {
        float h1v = fmaxf(c1[j]*sc1[j] + bi1[j], 0.f);
        myl[j*17 + m] = h1v;
      }
    }
    asm volatile("s_wait_dscnt 0" ::: "memory");
    float y = bias2;
    #pragma unroll
    for (int j = 0; j < 8; j++) y += w2v[j]*myl[j*17 + m];
    float sig = 1.f / (1.f + __expf(-y));
    vmax = fmaxf(vmax, sig);
  }
  if (lane < 16) vw_out[(size_t)(b*V_ + v)*HW_ + hw] = vmax;
}

// ---------------- k3: view-weighted combine + simnet MLP (WMMA) -> y0(B,D,H,W)
__global__ __launch_bounds__(256) void k_simnet(
    const float* __restrict__ sim,
    const float* __restrict__ vw,          // (B,V,H,W) from d_out
    const float* __restrict__ w0, const float* __restrict__ s0, const float* __restrict__ b0,
    const float* __restrict__ w1, const float* __restrict__ s1, const float* __restrict__ b1,
    const float* __restrict__ w2, const float* __restrict__ b2,
    float* __restrict__ y_out) {
  __shared__ float lds[8][272];
  int lane  = threadIdx.x & 31;
  int wslot = threadIdx.x >> 5;
  int wid   = blockIdx.x*8 + wslot;        // [0, PIX/16)
  int b     = wid / (HW_/16);
  int m     = lane & 15;
  int hi    = lane >> 4;
  int hw    = (wid % (HW_/16))*16 + m;

  v2f A0 = { w0[m*8 + 2*hi],     w0[m*8 + 1 + 2*hi] };
  v2f A1 = { w0[m*8 + 4 + 2*hi], w0[m*8 + 5 + 2*hi] };
  v2f A2[4];
  #pragma unroll
  for (int t = 0; t < 4; t++) {
    float axv = (m < 8) ? w1[m*16 + 4*t + 2*hi]     : 0.f;
    float ayv = (m < 8) ? w1[m*16 + 4*t + 1 + 2*hi] : 0.f;
    A2[t] = v2f{axv, ayv};
  }
  float sc0[8], bi0[8], sc1[8], bi1[8], w2v[8];
  #pragma unroll
  for (int j = 0; j < 8; j++) { sc0[j] = s0[j + 8*hi]; bi0[j] = b0[j + 8*hi]; }
  #pragma unroll
  for (int j = 0; j < 8; j++) { sc1[j] = s1[j]; bi1[j] = b1[j]; w2v[j] = w2[j]; }
  float bias2 = b2[0];

  float vwv[4], wsum = 0.f;
  #pragma unroll
  for (int v = 0; v < V_; v++) { float x = vw[(size_t)(b*V_ + v)*HW_ + hw]; vwv[v] = x; wsum += x; }
  float inw = 1.f / wsum;

  const size_t gs = (size_t)D_*HW_;
  float* myl = lds[wslot];
  const int goff[4] = {0, 1, 4, 5};

  for (int d = 0; d < D_; d++) {
    float xg[4];
    #pragma unroll
    for (int k = 0; k < 4; k++) {
      int g = 2*hi + goff[k];
      float a = 0.f;
      #pragma unroll
      for (int v = 0; v < V_; v++)
        a += sim[(size_t)(v*B_ + b)*G_*gs + (size_t)g*gs + (size_t)d*HW_ + hw] * vwv[v];
      xg[k] = a * inw;
    }
    v2f B0 = { xg[0], xg[1] };
    v2f B1 = { xg[2], xg[3] };
    v8f c0 = {};
    c0 = __builtin_amdgcn_wmma_f32_16x16x4_f32(false, A0, false, B0, (short)0, c0, false, false);
    c0 = __builtin_amdgcn_wmma_f32_16x16x4_f32(false, A1, false, B1, (short)0, c0, false, false);
    #pragma unroll
    for (int j = 0; j < 8; j++) {
      float h0v = fmaxf(c0[j]*sc0[j] + bi0[j], 0.f);
      myl[(j + 8*hi)*17 + m] = h0v;
    }
    asm volatile("s_wait_dscnt 0" ::: "memory");
    v8f c1 = {};
    #pragma unroll
    for (int t = 0; t < 4; t++) {
      v2f Bt = { myl[(4*t + 2*hi)*17 + m], myl[(4*t + 1 + 2*hi)*17 + m] };
      c1 = __builtin_amdgcn_wmma_f32_16x16x4_f32(false, A2[t], false, Bt, (short)0, c1, false, false);
    }
    if (hi == 0) {
      #pragma unroll
      for (int j = 0; j < 8; j++) {
        float h1v = fmaxf(c1[j]*sc1[j] + bi1[j], 0.f);
        myl[j*17 + m] = h1v;
      }
    }
    asm volatile("s_wait_dscnt 0" ::: "memory");
    float y = bias2;
    #pragma unroll
    for (int j = 0; j < 8; j++) y += w2v[j]*myl[j*17 + m];
    if (lane < 16) y_out[(size_t)(b*D_ + d)*HW_ + hw] = y;
  }
}

// ------------- k4: border-padded 9-neighbor bilinear gather * weight -> score
__global__ __launch_bounds__(256) void k_gridsample(
    const float* __restrict__ y0,     // (B,D,H,W)
    const float* __restrict__ grid,   // (B,NB*H,W,2)
    const float* __restrict__ wt,     // (B,D,NB,H,W)
    float* __restrict__ score_raw) {  // (B,D,H,W)
  int idx = blockIdx.x * blockDim.x + threadIdx.x;  // [0, POS)
  int hw = idx % HW_;
  int d  = (idx / HW_) % D_;
  int b  = idx / (D_*HW_);
  int w = hw % W_, h = hw / W_;
  const float* ybd = y0 + (size_t)(b*D_ + d)*HW_;
  float acc = 0.f;
  for (int nb = 0; nb < NB_; nb++) {
    const float* g = grid + ((size_t)(b*(NB_*H_) + nb*H_ + h)*W_ + w)*2;
    float gx = g[0], gy = g[1];
    // align_corners=False
    float ix = ((gx + 1.f)*(float)W_ - 1.f)*0.5f;
    float iy = ((gy + 1.f)*(float)H_ - 1.f)*0.5f;
    float x0f = floorf(ix), y0f = floorf(iy);
    float ax = ix - x0f, ay = iy - y0f;
    int x0i = (int)x0f, y0i = (int)y0f;
    int x0c = min(max(x0i, 0), W_-1), x1c = min(max(x0i+1, 0), W_-1);
    int y0c = min(max(y0i, 0), H_-1), y1c = min(max(y0i+1, 0), H_-1);
    float v00 = ybd[y0c*W_ + x0c], v01 = ybd[y0c*W_ + x1c];
    float v10 = ybd[y1c*W_ + x0c], v11 = ybd[y1c*W_ + x1c];
    float s = (1.f-ax)*(1.f-ay)*v00 + ax*(1.f-ay)*v01 + (1.f-ax)*ay*v10 + ax*ay*v11;
    acc += wt[(size_t)((b*D_ + d)*NB_ + nb)*HW_ + hw] * s;
  }
  score_raw[idx] = acc;
}

// --------------------------- k5: softmax over D + expected depth, final outputs
__global__ __launch_bounds__(256) void k_softmax_depth(
    const float* __restrict__ score_raw,
    const float* __restrict__ depth_sample,
    float* __restrict__ out_depth,
    float* __restrict__ out_score) {
  int p = blockIdx.x * blockDim.x + threadIdx.x;  // [0, PIX)
  int hw = p % HW_;
  int b  = p / HW_;
  const float* s  = score_raw    + (size_t)b*D_*HW_ + hw;
  const float* ds = depth_sample + (size_t)b*D_*HW_ + hw;
  float ev[D_];
  float mx = -1e30f;
  #pragma unroll
  for (int d = 0; d < D_; d++) { float x = s[(size_t)d*HW_]; ev[d] = x; mx = fmaxf(mx, x); }
  float sum = 0.f;
  #pragma unroll
  for (int d = 0; d < D_; d++) { float e = __expf(ev[d] - mx); ev[d] = e; sum += e; }
  float inv = 1.f / sum;
  float dep = 0.f;
  #pragma unroll
  for (int d = 0; d < D_; d++) {
    float sc = ev[d] * inv;
    out_score[(size_t)b*D_*HW_ + (size_t)d*HW_ + hw] = sc;
    dep += ds[(size_t)d*HW_] * sc;
  }
  out_depth[p] = dep;
}

// ---------------------------------------------------------------------- launch
extern "C" void kernel_launch(void* const* d_in, const int* in_sizes, int n_in,
                              void* d_out, int out_size, void* d_ws, size_t ws_size,
                              hipStream_t stream) {
  (void)in_sizes; (void)n_in; (void)out_size; (void)ws_size;
  const float* ref_feature  = (const float*)d_in[0];
  const float* src_features = (const float*)d_in[1];
  const float* ref_proj     = (const float*)d_in[2];
  const float* src_projs    = (const float*)d_in[3];
  const float* depth_sample = (const float*)d_in[4];
  const float* grid         = (const float*)d_in[7];
  const float* weight       = (const float*)d_in[8];
  const float* pw_w0 = (const float*)d_in[9];
  const float* pw_s0 = (const float*)d_in[10];
  const float* pw_b0 = (const float*)d_in[11];
  const float* pw_w1 = (const float*)d_in[12];
  const float* pw_s1 = (const float*)d_in[13];
  const float* pw_b1 = (const float*)d_in[14];
  const float* pw_w2 = (const float*)d_in[15];
  const float* pw_b2 = (const float*)d_in[16];
  const float* sn_w0 = (const float*)d_in[17];
  const float* sn_s0 = (const float*)d_in[18];
  const float* sn_b0 = (const float*)d_in[19];
  const float* sn_w1 = (const float*)d_in[20];
  const float* sn_s1 = (const float*)d_in[21];
  const float* sn_b1 = (const float*)d_in[22];
  const float* sn_w2 = (const float*)d_in[23];
  const float* sn_b2 = (const float*)d_in[24];

  float* out       = (float*)d_out;
  float* out_depth = out;                 // B*H*W
  float* out_score = out + PIX_;          // B*D*H*W
  float* out_vw    = out + PIX_ + POS_;   // B*V*H*W

  float* ws     = (float*)d_ws;
  float* ws_rot = ws + WS_ROT;
  float* ws_sim = ws + WS_SIM;
  float* ws_y0  = ws + WS_Y0;
  float* ws_sc  = ws + WS_SCORE;

  k_proj<<<1, 16, 0, stream>>>(ref_proj, src_projs, ws_rot);

  k_warp_sim<<<(V_*POS_)/256, 256, 0, stream>>>(
      ref_feature, src_features, depth_sample, ws_rot, ws_sim);

  k_pixelwise<<<(V_*(PIX_/16))/8, 256, 0, stream>>>(
      ws_sim, pw_w0, pw_s0, pw_b0, pw_w1, pw_s1, pw_b1, pw_w2, pw_b2, out_vw);

  k_simnet<<<((PIX_/16))/8, 256, 0, stream>>>(
      ws_sim, out_vw, sn_w0, sn_s0, sn_b0, sn_w1, sn_s1, sn_b1, sn_w2, sn_b2, ws_y0);

  k_gridsample<<<POS_/256, 256, 0, stream>>>(ws_y0, grid, weight, ws_sc);

  k_softmax_depth<<<PIX_/256, 256, 0, stream>>>(ws_sc, depth_sample, out_depth, out_score);
}